// NER_16578573762739
// MI455X (gfx1250) — compile-verified
//
#include <hip/hip_runtime.h>
#include <hip/hip_bf16.h>
#include <math.h>

typedef float v2f __attribute__((ext_vector_type(2)));
typedef float v8f __attribute__((ext_vector_type(8)));

#define T_LEN 256
#define B_SZ  512
#define E_DIM 300
#define H_DIM 100
#define K_TAGS 8
#define G4H   400   // 4*H

__device__ __forceinline__ float sigmoidf_(float x) { return 1.0f / (1.0f + __expf(-x)); }

// ---------------------------------------------------------------------------
// Input-projection GEMM:  out[m][n] = sum_k A[m][k] * W[n][k] + bih[n] + bhh[n]
// A is either direct (h1, row length Kd) or gathered rows of emb via `gather`.
// 256-thread block = 8 waves; each wave owns a 16-row strip. N=400 is split
// into 5 groups of 5 WMMA tiles so each A fragment feeds 5 v_wmma ops
// (5x fewer A-side loads than tile-at-a-time).
// ---------------------------------------------------------------------------
__global__ void proj_gemm(const float* __restrict__ A,
                          const int*   __restrict__ gather,
                          const float* __restrict__ emb,
                          int Kd,
                          const float* __restrict__ W,    // (400, Kd) row-major
                          const float* __restrict__ bih,
                          const float* __restrict__ bhh,
                          float* __restrict__ out)        // (M, 400)
{
    const int lane = threadIdx.x & 31;
    const int wv   = threadIdx.x >> 5;
    const int half = lane >> 4;
    const int ll   = lane & 15;
    const int m0   = blockIdx.x * 128 + wv * 16;

    const float* arow;
    if (gather) arow = emb + (size_t)gather[m0 + ll] * (size_t)Kd;
    else        arow = A   + (size_t)(m0 + ll) * (size_t)Kd;

    for (int ng = 0; ng < 5; ++ng) {
        const int nbase = ng * 80;                 // 5 tiles of 16 columns
        v8f acc[5];
#pragma unroll
        for (int j = 0; j < 5; ++j) {
            const float bias = bih[nbase + j * 16 + ll] + bhh[nbase + j * 16 + ll];
#pragma unroll
            for (int v = 0; v < 8; ++v) acc[j][v] = bias;
        }
        for (int k = 0; k < Kd; k += 4) {
            v2f a;
            a.x = arow[k + 2 * half];
            a.y = arow[k + 2 * half + 1];
#pragma unroll
            for (int j = 0; j < 5; ++j) {
                const float* wrow = W + (size_t)(nbase + j * 16 + ll) * (size_t)Kd
                                      + k + 2 * half;
                v2f b;
                b.x = wrow[0];
                b.y = wrow[1];
                acc[j] = __builtin_amdgcn_wmma_f32_16x16x4_f32(false, a, false, b,
                                                               (short)0, acc[j],
                                                               false, false);
            }
        }
#pragma unroll
        for (int j = 0; j < 5; ++j)
#pragma unroll
            for (int v = 0; v < 8; ++v)
                out[(size_t)(m0 + v + 8 * half) * G4H + nbase + j * 16 + ll] = acc[j][v];
    }
}

// ---------------------------------------------------------------------------
// LSTM recurrent scan, both directions (blockIdx.y). Each block owns a 16-row
// batch tile; h/c state lives in LDS for the whole time loop.
// The next timestep's xp tile (16x400 = 25.6KB) is prefetched into LDS with
// GLOBAL_LOAD_ASYNC_TO_LDS_B32 (ASYNCcnt) while the current step's WMMA runs;
// the consumed xp buffer is reused in-place as the gate buffer g.
// ---------------------------------------------------------------------------
__global__ void lstm_scan(const float* __restrict__ xp_f,
                          const float* __restrict__ xp_b,
                          const float* __restrict__ whh_f,  // (400, 100)
                          const float* __restrict__ whh_b,
                          float* __restrict__ hout)         // (T, B, 2H)
{
    const int dir = blockIdx.y;
    const float* xp  = dir ? xp_b  : xp_f;
    const float* whh = dir ? whh_b : whh_f;
    const int b0 = blockIdx.x * 16;

    __shared__ float xp_s[2][16 * G4H];   // 51,200 B : double-buffered xp / g
    __shared__ float h_s[16 * 104];       //  6,656 B
    __shared__ float c_s[16 * 104];       //  6,656 B   (total 64,512 B)

    const int tid  = threadIdx.x;
    const int lane = tid & 31;
    const int wv   = tid >> 5;
    const int half = lane >> 4;
    const int ll   = lane & 15;

    // Pull the 160KB recurrent weight matrix toward L2/WGP$ before the scan.
    for (int i = tid; i < 1250; i += 256)
        __builtin_prefetch(whh + (size_t)i * 32, 0, 1);

    for (int i = tid; i < 16 * 104; i += 256) { h_s[i] = 0.0f; c_s[i] = 0.0f; }

    // Async-copy one 16x400 tile (6400 floats, contiguous) into xp_s[buf].
    // 25 B32 ops per thread -> 25 ASYNCcnt increments per wave.
    auto issue_tile = [&](int tt, int buf) {
        const float* src = xp + ((size_t)tt * B_SZ + b0) * G4H;
        const unsigned ldsbase = (unsigned)(size_t)(&xp_s[buf][0]);
#pragma unroll
        for (int k = 0; k < 25; ++k) {
            const int e = tid + k * 256;
            const float* gp = src + e;
            const unsigned lo = ldsbase + 4u * (unsigned)e;
            asm volatile("global_load_async_to_lds_b32 %0, %1, off"
                         :: "v"(lo), "v"(gp) : "memory");
        }
    };

    issue_tile(dir ? (T_LEN - 1) : 0, 0);

    for (int t = 0; t < T_LEN; ++t) {
        const int tt  = dir ? (T_LEN - 1 - t) : t;
        const int cur = t & 1;

        if (t + 1 < T_LEN) {
            issue_tile(dir ? (T_LEN - 2 - t) : (t + 1), cur ^ 1);
            // <=25 outstanding => the 25 ops for step t have landed (in-order)
            asm volatile("s_wait_asynccnt 25" ::: "memory");
        } else {
            asm volatile("s_wait_asynccnt 0" ::: "memory");
        }
        __syncthreads();   // all waves' async tiles visible; h_s/c_s init done

        float* gbuf = &xp_s[cur][0];

        // Phase 1: g = xp[tt] + h @ whh^T via WMMA; g overwrites xp in-place
        for (int nt = wv; nt < 25; nt += 8) {
            const int n0 = nt * 16;
            v8f acc;
#pragma unroll
            for (int v = 0; v < 8; ++v)
                acc[v] = gbuf[(v + 8 * half) * G4H + n0 + ll];
            for (int k = 0; k < H_DIM; k += 4) {
                v2f a, b;
                a.x = h_s[ll * 104 + k + 2 * half];
                a.y = h_s[ll * 104 + k + 2 * half + 1];
                const float* wrow = whh + (size_t)(n0 + ll) * H_DIM + k + 2 * half;
                b.x = wrow[0];
                b.y = wrow[1];
                acc = __builtin_amdgcn_wmma_f32_16x16x4_f32(false, a, false, b,
                                                            (short)0, acc,
                                                            false, false);
            }
#pragma unroll
            for (int v = 0; v < 8; ++v)
                gbuf[(v + 8 * half) * G4H + n0 + ll] = acc[v];
        }
        __syncthreads();

        // Phase 2: gates + state update (16*100 elements over 256 threads)
        for (int idx = tid; idx < 16 * H_DIM; idx += 256) {
            const int bl = idx / H_DIM;
            const int j  = idx - bl * H_DIM;
            const float ig = sigmoidf_(gbuf[bl * G4H + j]);
            const float fg = sigmoidf_(gbuf[bl * G4H + H_DIM + j]);
            const float gg = tanhf    (gbuf[bl * G4H + 2 * H_DIM + j]);
            const float og = sigmoidf_(gbuf[bl * G4H + 3 * H_DIM + j]);
            const float c  = fg * c_s[bl * 104 + j] + ig * gg;
            c_s[bl * 104 + j] = c;
            const float h  = og * tanhf(c);
            h_s[bl * 104 + j] = h;
            hout[((size_t)tt * B_SZ + b0 + bl) * (2 * H_DIM) + dir * H_DIM + j] = h;
        }
        __syncthreads();   // h_s stable for next phase 1; old buffer now free
    }
}

// ---------------------------------------------------------------------------
// Linear head: one thread per row m computes all 8 classes; wl staged in LDS.
// ---------------------------------------------------------------------------
__global__ void linear_em(const float* __restrict__ h2,
                          const float* __restrict__ wl,   // (8, 200)
                          const float* __restrict__ bl,
                          float* __restrict__ em)         // (M, 8)
{
    __shared__ float wls[K_TAGS * 200];
    __shared__ float bls[K_TAGS];
    const int tid = threadIdx.x;
    for (int i = tid; i < K_TAGS * 200; i += 256) wls[i] = wl[i];
    if (tid < K_TAGS) bls[tid] = bl[tid];
    __syncthreads();

    const size_t m = (size_t)blockIdx.x * 256 + tid;
    float s[K_TAGS];
#pragma unroll
    for (int k = 0; k < K_TAGS; ++k) s[k] = bls[k];
    const float* hrow = h2 + m * (2 * H_DIM);
    for (int j = 0; j < 2 * H_DIM; ++j) {
        const float hv = hrow[j];
#pragma unroll
        for (int k = 0; k < K_TAGS; ++k) s[k] += hv * wls[k * 200 + j];
    }
#pragma unroll
    for (int k = 0; k < K_TAGS; ++k) em[m * K_TAGS + k] = s[k];
}

// ---------------------------------------------------------------------------
// CRF log-likelihood: one thread per batch element, K=8 states.
// Block-reduced partial sums, atomicAdd into loss (pre-zeroed).
// ---------------------------------------------------------------------------
__global__ void crf_loss(const float* __restrict__ em,
                         const int*   __restrict__ tags,
                         const unsigned char* __restrict__ mask,
                         const float* __restrict__ start_t,
                         const float* __restrict__ end_t,
                         const float* __restrict__ trans,
                         float* __restrict__ loss)
{
    __shared__ float tr[64];
    __shared__ float red[256];
    const int tid = threadIdx.x;
    if (tid < 64) tr[tid] = trans[tid];
    __syncthreads();

    const int b = blockIdx.x * 256 + tid;
    float partial = 0.0f;
    if (b < B_SZ) {
        float score[K_TAGS];
        const int tag0 = tags[b];
        float num = start_t[tag0] + em[(size_t)b * K_TAGS + tag0];
#pragma unroll
        for (int j = 0; j < 8; ++j) score[j] = start_t[j] + em[(size_t)b * K_TAGS + j];

        int prev = tag0;
        int msum = mask[b] ? 1 : 0;
        for (int t = 1; t < T_LEN; ++t) {
            const float* emt = em + ((size_t)t * B_SZ + b) * K_TAGS;
            const float m = mask[t * B_SZ + b] ? 1.0f : 0.0f;
            msum += (int)m;
            const int tg = tags[t * B_SZ + b];
            num += (tr[prev * 8 + tg] + emt[tg]) * m;
            prev = tg;
            float nxt[8];
#pragma unroll
            for (int j = 0; j < 8; ++j) {
                float mx = -1e30f;
#pragma unroll
                for (int i = 0; i < 8; ++i) mx = fmaxf(mx, score[i] + tr[i * 8 + j]);
                float s = 0.0f;
#pragma unroll
                for (int i = 0; i < 8; ++i) s += __expf(score[i] + tr[i * 8 + j] - mx);
                nxt[j] = mx + __logf(s) + emt[j];
            }
#pragma unroll
            for (int j = 0; j < 8; ++j) score[j] = (m > 0.0f) ? nxt[j] : score[j];
        }
        const int send = msum - 1;
        num += end_t[tags[(size_t)send * B_SZ + b]];
        float mx = -1e30f;
#pragma unroll
        for (int j = 0; j < 8; ++j) mx = fmaxf(mx, score[j] + end_t[j]);
        float s = 0.0f;
#pragma unroll
        for (int j = 0; j < 8; ++j) s += __expf(score[j] + end_t[j] - mx);
        partial = num - (mx + __logf(s));
    }
    red[tid] = partial;
    __syncthreads();
    for (int st = 128; st > 0; st >>= 1) {
        if (tid < st) red[tid] += red[tid + st];
        __syncthreads();
    }
    if (tid == 0) atomicAdd(loss, red[0]);
}

// ---------------------------------------------------------------------------
// Viterbi decode: forward max-product with argmax history, then backtrace.
// ---------------------------------------------------------------------------
__global__ void viterbi(const float* __restrict__ em,
                        const float* __restrict__ start_t,
                        const float* __restrict__ end_t,
                        const float* __restrict__ trans,
                        int* __restrict__ hist,    // (T-1, B, 8)
                        float* __restrict__ dec)   // (T, B)
{
    __shared__ float tr[64];
    const int tid = threadIdx.x;
    if (tid < 64) tr[tid] = trans[tid];
    __syncthreads();

    const int b = blockIdx.x * 256 + tid;
    if (b >= B_SZ) return;

    float score[8];
#pragma unroll
    for (int j = 0; j < 8; ++j) score[j] = start_t[j] + em[(size_t)b * 8 + j];

    for (int t = 1; t < T_LEN; ++t) {
        const float* emt = em + ((size_t)t * B_SZ + b) * 8;
        float ns[8]; int ni[8];
#pragma unroll
        for (int j = 0; j < 8; ++j) {
            float best = score[0] + tr[j];
            int bi = 0;
#pragma unroll
            for (int i = 1; i < 8; ++i) {
                const float v = score[i] + tr[i * 8 + j];
                if (v > best) { best = v; bi = i; }
            }
            ns[j] = best + emt[j];
            ni[j] = bi;
        }
#pragma unroll
        for (int j = 0; j < 8; ++j) {
            score[j] = ns[j];
            hist[((size_t)(t - 1) * B_SZ + b) * 8 + j] = ni[j];
        }
    }
    int last = 0;
    float best = score[0] + end_t[0];
#pragma unroll
    for (int j = 1; j < 8; ++j) {
        const float v = score[j] + end_t[j];
        if (v > best) { best = v; last = j; }
    }
    dec[(size_t)(T_LEN - 1) * B_SZ + b] = (float)last;
    int nxt = last;
    for (int t = T_LEN - 2; t >= 0; --t) {
        const int prev = hist[((size_t)t * B_SZ + b) * 8 + nxt];
        dec[(size_t)t * B_SZ + b] = (float)prev;
        nxt = prev;
    }
}

// ---------------------------------------------------------------------------
extern "C" void kernel_launch(void* const* d_in, const int* in_sizes, int n_in,
                              void* d_out, int out_size, void* d_ws, size_t ws_size,
                              hipStream_t stream)
{
    (void)in_sizes; (void)n_in; (void)out_size; (void)ws_size;

    const int*   sentence = (const int*)  d_in[0];
    const int*   tags     = (const int*)  d_in[1];
    const unsigned char* mask = (const unsigned char*)d_in[2];
    const float* emb   = (const float*)d_in[3];
    const float* wih0f = (const float*)d_in[4];
    const float* whh0f = (const float*)d_in[5];
    const float* bih0f = (const float*)d_in[6];
    const float* bhh0f = (const float*)d_in[7];
    const float* wih0b = (const float*)d_in[8];
    const float* whh0b = (const float*)d_in[9];
    const float* bih0b = (const float*)d_in[10];
    const float* bhh0b = (const float*)d_in[11];
    const float* wih1f = (const float*)d_in[12];
    const float* whh1f = (const float*)d_in[13];
    const float* bih1f = (const float*)d_in[14];
    const float* bhh1f = (const float*)d_in[15];
    const float* wih1b = (const float*)d_in[16];
    const float* whh1b = (const float*)d_in[17];
    const float* bih1b = (const float*)d_in[18];
    const float* bhh1b = (const float*)d_in[19];
    const float* wl      = (const float*)d_in[20];
    const float* bl      = (const float*)d_in[21];
    const float* start_t = (const float*)d_in[22];
    const float* end_t   = (const float*)d_in[23];
    const float* trans   = (const float*)d_in[24];

    float* out = (float*)d_out;
    char*  ws  = (char*)d_ws;
    const size_t M = (size_t)T_LEN * B_SZ;

    size_t off = 0;
    float* xp_f = (float*)(ws + off); off += M * G4H * sizeof(float);
    float* xp_b = (float*)(ws + off); off += M * G4H * sizeof(float);
    float* h1   = (float*)(ws + off); off += M * (2 * H_DIM) * sizeof(float);
    float* h2   = (float*)(ws + off); off += M * (2 * H_DIM) * sizeof(float);
    float* em   = (float*)(ws + off); off += M * K_TAGS * sizeof(float);
    int*   hist = (int*)  (ws + off); off += M * K_TAGS * sizeof(int);

    hipMemsetAsync(d_out, 0, sizeof(float), stream);  // loss accumulator

    const dim3 blk(256);
    const dim3 gproj((unsigned)(M / 128));            // 1024 blocks, 8 waves each
    const dim3 gscan(B_SZ / 16, 2);                   // 32 batch tiles x 2 dirs

    // Layer 0: embedding-gather fused projection, then recurrent scan
    proj_gemm<<<gproj, blk, 0, stream>>>(nullptr, sentence, emb, E_DIM,
                                         wih0f, bih0f, bhh0f, xp_f);
    proj_gemm<<<gproj, blk, 0, stream>>>(nullptr, sentence, emb, E_DIM,
                                         wih0b, bih0b, bhh0b, xp_b);
    lstm_scan<<<gscan, blk, 0, stream>>>(xp_f, xp_b, whh0f, whh0b, h1);

    // Layer 1
    proj_gemm<<<gproj, blk, 0, stream>>>(h1, nullptr, nullptr, 2 * H_DIM,
                                         wih1f, bih1f, bhh1f, xp_f);
    proj_gemm<<<gproj, blk, 0, stream>>>(h1, nullptr, nullptr, 2 * H_DIM,
                                         wih1b, bih1b, bhh1b, xp_b);
    lstm_scan<<<gscan, blk, 0, stream>>>(xp_f, xp_b, whh1f, whh1b, h2);

    // Emissions + CRF loss + Viterbi decode
    linear_em<<<dim3((unsigned)(M / 256)), blk, 0, stream>>>(h2, wl, bl, em);
    crf_loss<<<dim3(2), blk, 0, stream>>>(em, tags, mask, start_t, end_t, trans, out);
    viterbi<<<dim3(2), blk, 0, stream>>>(em, start_t, end_t, trans, hist, out + 1);
}